// QuantumAdaGN_80736795230973
// MI455X (gfx1250) — compile-verified
//
#include <hip/hip_runtime.h>
#include <hip/hip_bf16.h>

// ---------------------------------------------------------------------------
// QuantumAdaGN fused kernel for MI455X (gfx1250, wave32).
//   traffic: 33.5MB in + 33.5MB out  -> ~2.9us at 23.3 TB/s (memory roof)
//   compute: ~2.1 GFLOP via v_wmma_f32_16x16x4_f32 + ~1 GFLOP VALU circuit
// One wave = 16 pixels. Quantum state kept entirely in the WMMA D layout:
//   q = (lane&15) + 16*tile, pixel = j + 8*(lane>>4)   (tile=0..3, j=0..7)
// qubit pair bits 32/16 -> register renames; bits 8/4/2/1 -> ds_swizzle XOR.
// W_in/W_out staged in LDS in *fragment-packed* [frag][lane][2] layout so
// every WMMA B operand is a single aligned ds_load_b64 (no repacking movs).
// ---------------------------------------------------------------------------

typedef __attribute__((ext_vector_type(2))) float v2f;
typedef __attribute__((ext_vector_type(8))) float v8f;

#define C_DIM 128
#define Q_DIM 64
#define HW    4096
#define PP    68      // probs staging pitch (avoids 16-way bank conflict)
#define OP    20      // out staging pitch  (float4-aligned, few conflicts)
#define STG_F 2560    // per-wave staging floats (>= 128*OP)

__device__ __forceinline__ v8f wmma4(v2f a, v2f b, v8f c) {
  // D = A(16x4,f32) * B(4x16,f32) + C(16x16,f32)
  return __builtin_amdgcn_wmma_f32_16x16x4_f32(
      false, a, false, b, (short)0, c, false, false);
}

template <int XORM>
__device__ __forceinline__ float lane_xor(float v) {
  // ds_swizzle_b32 group-of-32 mode: new_lane = ((lane & and) | or) ^ xor
  // offset = {xor[14:10], or[9:5], and[4:0]}
  int i = __builtin_amdgcn_ds_swizzle(__float_as_int(v), (XORM << 10) | 0x1f);
  return __int_as_float(i);
}

// Rotation with pairing across accumulator tiles (qubit bit 32 -> TBIT=2,
// qubit bit 16 -> TBIT=1). Pure register work.
template <int TBIT, bool RX>
__device__ __forceinline__ void rot_tile(float sr[4][8], float si[4][8],
                                         float ch, float sh) {
#pragma unroll
  for (int t = 0; t < 4; ++t) {
    if (t & TBIT) continue;
    const int u = t | TBIT;
#pragma unroll
    for (int j = 0; j < 8; ++j) {
      float r0 = sr[t][j], i0 = si[t][j];
      float r1 = sr[u][j], i1 = si[u][j];
      if constexpr (RX) {  // b=c=-i*sh
        sr[t][j] = ch * r0 + sh * i1;  si[t][j] = ch * i0 - sh * r1;
        sr[u][j] = ch * r1 + sh * i0;  si[u][j] = ch * i1 - sh * r0;
      } else {             // b=-sh, c=+sh (real)
        sr[t][j] = ch * r0 - sh * r1;  si[t][j] = ch * i0 - sh * i1;
        sr[u][j] = ch * r1 + sh * r0;  si[u][j] = ch * i1 + sh * i0;
      }
    }
  }
}

// Rotation with pairing across lanes (qubit bits 8/4/2/1 -> P).
template <int P, bool RX>
__device__ __forceinline__ void rot_lane(float sr[4][8], float si[4][8],
                                         float ch, float sh, int lane) {
  const float sgn = (lane & P) ? sh : -sh;  // RY sign by pair parity
#pragma unroll
  for (int t = 0; t < 4; ++t) {
#pragma unroll
    for (int j = 0; j < 8; ++j) {
      float r = sr[t][j], i = si[t][j];
      float pr = lane_xor<P>(r);
      float pi = lane_xor<P>(i);
      if constexpr (RX) {  // symmetric in pair parity
        sr[t][j] = ch * r + sh * pi;
        si[t][j] = ch * i - sh * pr;
      } else {
        sr[t][j] = ch * r + sgn * pr;
        si[t][j] = ch * i + sgn * pi;
      }
    }
  }
}

// CNOT with control lane-bit CB and target lane-bit TB (swizzle + cndmask).
template <int CB, int TB>
__device__ __forceinline__ void cnot_lane(float sr[4][8], float si[4][8],
                                          int lane) {
  const bool sel = (lane & CB) != 0;
#pragma unroll
  for (int t = 0; t < 4; ++t) {
#pragma unroll
    for (int j = 0; j < 8; ++j) {
      float vr = lane_xor<TB>(sr[t][j]);
      float vi = lane_xor<TB>(si[t][j]);
      sr[t][j] = sel ? vr : sr[t][j];
      si[t][j] = sel ? vi : si[t][j];
    }
  }
}

// ---------------------------------------------------------------------------
// Kernel 1: per-batch angle table.  tab[b][j] = (cos(h), sin(h)),
// h = 0.5*(weights[j] + style[b] @ W_style[:,j] + b_style[j]);  16*18 entries.
// ---------------------------------------------------------------------------
extern "C" __global__ void qadagn_angles(const float* __restrict__ style,
                                         const float* __restrict__ W_style,
                                         const float* __restrict__ b_style,
                                         const float* __restrict__ weights,
                                         float* __restrict__ tab) {
  int id = blockIdx.x * blockDim.x + threadIdx.x;
  if (id >= 16 * 18) return;
  int b = id / 18, j = id % 18;
  const float* s = style + b * 512;
  float acc = b_style[j] + weights[j];
  for (int k = 0; k < 512; ++k) acc += s[k] * W_style[k * 18 + j];
  float h = 0.5f * acc;
  tab[id * 2 + 0] = cosf(h);
  tab[id * 2 + 1] = sinf(h);
}

// ---------------------------------------------------------------------------
// Kernel 2: fused GEMM1 -> normalize -> circuit -> |.|^2 -> GEMM2 -> store.
// block = 128 threads = 4 waves; each wave owns 16 consecutive pixels.
// ---------------------------------------------------------------------------
extern "C" __global__ void __launch_bounds__(128)
qadagn_main(const float* __restrict__ x, const float* __restrict__ W_in,
            const float* __restrict__ b_in, const float* __restrict__ W_out,
            const float* __restrict__ b_out, const float* __restrict__ tab,
            float* __restrict__ out) {
  extern __shared__ float smem[];
  float* sWin  = smem;             // 128 frags * 64 floats = 8192
  float* sWout = smem + 8192;      // 128 frags * 64 floats = 8192
  const int tid  = threadIdx.x;
  const int lane = tid & 31;
  const int wv   = tid >> 5;
  float* stg = smem + 16384 + wv * STG_F;  // per-wave scratch

  // ---- stage weights in fragment-packed layout: [frag][lane][2] ----------
  // W_in frag f = (k0/4)*4 + t : pair c holds W_in[k0 + (l>>4)*2 + c][t*16+l&15]
  for (int i = tid; i < 8192; i += 128) {
    const int c = i & 1;
    const int l = (i >> 1) & 31;
    const int f = i >> 6;
    const int row = ((l >> 4) << 1) + c;
    const int col = (l & 15);
    {  // W_in: f = (k0>>2)*4 + t
      const int k0 = (f >> 2) << 2;
      const int t  = f & 3;
      sWin[i] = W_in[(k0 + row) * Q_DIM + t * 16 + col];
    }
    {  // W_out: f = (k0>>2)*8 + n
      const int k0 = (f >> 3) << 2;
      const int n  = f & 7;
      sWout[i] = W_out[(k0 + row) * C_DIM + n * 16 + col];
    }
  }
  __syncthreads();

  const int p0  = blockIdx.x * 64 + wv * 16;  // tile base pixel
  const int b   = p0 >> 12;                   // image index (HW=4096 | 16)
  const int hw  = p0 & 4095;
  const int lo  = lane & 15;
  const int hi  = lane >> 4;   // which half (pixel row group)
  const int hi2 = hi << 1;     // K sub-offset for f32 WMMA A/B frags

  const float* wfin  = sWin + lane * 2;   // packed B-frag base (8B aligned)
  const float* wfout = sWout + lane * 2;

  // ---------------- GEMM1: x_q = x_tile @ W_in  (16x128 * 128x64) ----------
  const float* xb = x + (size_t)b * C_DIM * HW + hw + lo;
  v8f acc[4] = {};
#pragma unroll
  for (int k0 = 0; k0 < C_DIM; k0 += 4) {
    v2f a;
    a.x = xb[(k0 + hi2) * HW];        // A[m=lo][k0 + hi2]
    a.y = xb[(k0 + hi2 + 1) * HW];    // A[m=lo][k0 + hi2 + 1]
#pragma unroll
    for (int t = 0; t < 4; ++t) {
      v2f bb = *(const v2f*)(wfin + ((k0 >> 2) * 4 + t) * 64);  // ds_load_b64
      acc[t] = wmma4(a, bb, acc[t]);
    }
  }

  // +b_in, init complex state
  float sr[4][8], si[4][8];
#pragma unroll
  for (int t = 0; t < 4; ++t) {
    const float bi = b_in[t * 16 + lo];
#pragma unroll
    for (int j = 0; j < 8; ++j) { sr[t][j] = acc[t][j] + bi; si[t][j] = 0.f; }
  }

  // ---------------- per-pixel L2 normalize (16-lane XOR butterfly) ---------
#pragma unroll
  for (int j = 0; j < 8; ++j) {
    float s = 0.f;
#pragma unroll
    for (int t = 0; t < 4; ++t) s += sr[t][j] * sr[t][j];
    s += lane_xor<1>(s);
    s += lane_xor<2>(s);
    s += lane_xor<4>(s);
    s += lane_xor<8>(s);
    const float inv = 1.0f / (sqrtf(s) + 1e-9f);
#pragma unroll
    for (int t = 0; t < 4; ++t) sr[t][j] *= inv;
  }

  // ---------------- quantum circuit (angles uniform per batch) -------------
  const float* ab = tab + b * 36;  // [wire*3+k][{cos,sin}]
#define CH(w, k) ab[((w) * 3 + (k)) * 2]
#define SH(w, k) ab[((w) * 3 + (k)) * 2 + 1]
  // layer: RX then RY on each wire
  rot_tile<2, true >(sr, si, CH(0, 0), SH(0, 0));
  rot_tile<2, false>(sr, si, CH(0, 1), SH(0, 1));
  rot_tile<1, true >(sr, si, CH(1, 0), SH(1, 0));
  rot_tile<1, false>(sr, si, CH(1, 1), SH(1, 1));
  rot_lane<8, true >(sr, si, CH(2, 0), SH(2, 0), lane);
  rot_lane<8, false>(sr, si, CH(2, 1), SH(2, 1), lane);
  rot_lane<4, true >(sr, si, CH(3, 0), SH(3, 0), lane);
  rot_lane<4, false>(sr, si, CH(3, 1), SH(3, 1), lane);
  rot_lane<2, true >(sr, si, CH(4, 0), SH(4, 0), lane);
  rot_lane<2, false>(sr, si, CH(4, 1), SH(4, 1), lane);
  rot_lane<1, true >(sr, si, CH(5, 0), SH(5, 0), lane);
  rot_lane<1, false>(sr, si, CH(5, 1), SH(5, 1), lane);
  // CNOT chain ctrl=0..4 (control bit 32>>c, target bit 16>>c)
#pragma unroll
  for (int j = 0; j < 8; ++j) {  // ctrl=0: tiles 2<->3
    float tr = sr[2][j]; sr[2][j] = sr[3][j]; sr[3][j] = tr;
    float ti = si[2][j]; si[2][j] = si[3][j]; si[3][j] = ti;
  }
#pragma unroll
  for (int j = 0; j < 8; ++j) {  // ctrl=1: tiles 1,3 full lane-xor-8
    sr[1][j] = lane_xor<8>(sr[1][j]); si[1][j] = lane_xor<8>(si[1][j]);
    sr[3][j] = lane_xor<8>(sr[3][j]); si[3][j] = lane_xor<8>(si[3][j]);
  }
  cnot_lane<8, 4>(sr, si, lane);  // ctrl=2
  cnot_lane<4, 2>(sr, si, lane);  // ctrl=3
  cnot_lane<2, 1>(sr, si, lane);  // ctrl=4
  // final RY layer
  rot_tile<2, false>(sr, si, CH(0, 2), SH(0, 2));
  rot_tile<1, false>(sr, si, CH(1, 2), SH(1, 2));
  rot_lane<8, false>(sr, si, CH(2, 2), SH(2, 2), lane);
  rot_lane<4, false>(sr, si, CH(3, 2), SH(3, 2), lane);
  rot_lane<2, false>(sr, si, CH(4, 2), SH(4, 2), lane);
  rot_lane<1, false>(sr, si, CH(5, 2), SH(5, 2), lane);
#undef CH
#undef SH

  // ---------------- probs -> LDS (D layout -> A layout transpose) ----------
#pragma unroll
  for (int t = 0; t < 4; ++t)
#pragma unroll
    for (int j = 0; j < 8; ++j) {
      float p = sr[t][j] * sr[t][j] + si[t][j] * si[t][j];
      stg[(j + 8 * hi) * PP + (lo + 16 * t)] = p;  // [pixel][q]
    }
  __syncthreads();

  // ---------------- GEMM2: out = probs @ W_out  (16x64 * 64x128) -----------
  v8f oacc[8] = {};
#pragma unroll
  for (int k0 = 0; k0 < Q_DIM; k0 += 4) {
    // rows k0+hi2, k0+hi2+1 adjacent in staging -> single ds_load_b64
    v2f a = *(const v2f*)(stg + lo * PP + k0 + hi2);
#pragma unroll
    for (int n = 0; n < 8; ++n) {
      v2f bb = *(const v2f*)(wfout + ((k0 >> 2) * 8 + n) * 64);  // ds_load_b64
      oacc[n] = wmma4(a, bb, oacc[n]);
    }
  }
  __syncthreads();

  // +b_out, stage as [channel][pixel] so global stores are pixel-contiguous
#pragma unroll
  for (int n = 0; n < 8; ++n) {
    const float bo = b_out[n * 16 + lo];
#pragma unroll
    for (int j = 0; j < 8; ++j)
      stg[(n * 16 + lo) * OP + (j + 8 * hi)] = oacc[n][j] + bo;
  }
  __syncthreads();

  // coalesced float4 stores: out[b, c, hw + p]
  float* ob = out + (size_t)b * C_DIM * HW + hw;
#pragma unroll
  for (int cc = 0; cc < 16; ++cc) {
    const int c  = cc * 8 + (lane >> 2);
    const int p4 = (lane & 3) * 4;
    float4 v = *(const float4*)(stg + c * OP + p4);
    *(float4*)(ob + c * HW + p4) = v;
  }
}

// ---------------------------------------------------------------------------
extern "C" void kernel_launch(void* const* d_in, const int* in_sizes, int n_in,
                              void* d_out, int out_size, void* d_ws,
                              size_t ws_size, hipStream_t stream) {
  const float* x       = (const float*)d_in[0];
  const float* style   = (const float*)d_in[1];
  const float* W_in    = (const float*)d_in[2];
  const float* b_in    = (const float*)d_in[3];
  const float* W_style = (const float*)d_in[4];
  const float* b_style = (const float*)d_in[5];
  const float* weights = (const float*)d_in[6];
  const float* W_out   = (const float*)d_in[7];
  const float* b_out   = (const float*)d_in[8];
  float* out = (float*)d_out;
  float* tab = (float*)d_ws;  // 16*18*2 floats

  qadagn_angles<<<9, 32, 0, stream>>>(style, W_style, b_style, weights, tab);

  const size_t smem = (16384 + 4 * STG_F) * sizeof(float);  // ~104 KB
  // 65536 pixels / (4 waves * 16 px) = 1024 workgroups
  qadagn_main<<<1024, 128, smem, stream>>>(x, W_in, b_in, W_out, b_out, tab,
                                           out);
}